// FPSPoolLayer_16484084483004
// MI455X (gfx1250) — compile-verified
//
#include <hip/hip_runtime.h>
#include <stdint.h>

// Problem constants (match reference)
#define B_CLOUDS 16
#define NPTS     16384
#define KSAMP    4096
#define TPB      512                 // 16 waves of 32
#define NWAVES   (TPB / 32)          // 16
#define PPT      (NPTS / TPB)        // 32 points per thread, held in registers

// One workgroup per cloud. Entire cloud (192KB) lives in LDS (320KB/WGP on CDNA5).
// Per-thread state (coords + running min-dist) lives in VGPRs.
// Per round: 1 workgroup barrier; wave32 shuffle argmax reduction; cross-wave
// combine via a parity double-buffered per-wave-best array that each wave
// re-reduces lane-parallel (1 load + 4 shuffle steps). Initial stage-in uses
// gfx1250 async global->LDS DMA tracked by ASYNCcnt.
__global__ __launch_bounds__(TPB) void fps_kernel(const float* __restrict__ x,
                                                  float* __restrict__ xout) {
    __shared__ float pts[NPTS * 3];                       // 196608 bytes
    __shared__ unsigned long long wbest[2][NWAVES];       // parity double buffer

    const int b = blockIdx.x;
    const int t = threadIdx.x;
    const float* src = x + (size_t)b * NPTS * 3;

    // ---- Stage cloud into LDS with CDNA5 async global->LDS B128 copies ----
    {
        unsigned ldsBase = (unsigned)(uintptr_t)(&pts[0]);   // LDS byte offset
        uint64_t gBase   = (uint64_t)(uintptr_t)src;
        const int chunks = (NPTS * 3 * 4) / 16;              // 12288 x 16B
        for (int c = t; c < chunks; c += TPB) {
            unsigned laddr = ldsBase + (unsigned)c * 16u;
            uint64_t gaddr = gBase + (uint64_t)c * 16u;
            asm volatile("global_load_async_to_lds_b128 %0, %1, off"
                         :: "v"(laddr), "v"(gaddr) : "memory");
        }
        asm volatile("s_wait_asynccnt 0" ::: "memory");
    }
    __syncthreads();

    // ---- Pull owned points into registers (stride-TPB ownership: conflict-free) ----
    float px[PPT], py[PPT], pz[PPT], mind[PPT];
#pragma unroll
    for (int i = 0; i < PPT; ++i) {
        const int g = i * TPB + t;
        px[i] = pts[3 * g + 0];
        py[i] = pts[3 * g + 1];
        pz[i] = pts[3 * g + 2];
        mind[i] = __FLT_MAX__;       // round 1 folds in the seed distance
    }

    // Seed: sample 0 is point 0 of the cloud.
    float wx = pts[0], wy = pts[1], wz = pts[2];
    if (t == 0) {
        float* o = xout + (size_t)b * KSAMP * 3;
        o[0] = wx; o[1] = wy; o[2] = wz;
    }

    for (int k = 1; k < KSAMP; ++k) {
        // Update mind vs previous winner and track local argmax (first-index ties).
        float bestd = -1.0f;
        unsigned besti = 0u;
#pragma unroll
        for (int i = 0; i < PPT; ++i) {
            const float dx = px[i] - wx;
            const float dy = py[i] - wy;
            const float dz = pz[i] - wz;
            float d = dx * dx + dy * dy + dz * dz;
            float m = fminf(mind[i], d);
            mind[i] = m;
            if (m > bestd) { bestd = m; besti = (unsigned)(i * TPB + t); }
        }
        // Pack: high 32 = float bits (monotonic for >=0), low 32 = ~index so that
        // u64-max ties break toward the SMALLEST global index (matches jnp.argmax).
        unsigned long long key =
            ((unsigned long long)__float_as_uint(bestd) << 32) |
            (unsigned long long)(0xFFFFFFFFu - besti);

        // Wave32 max-reduction (ds_bpermute-based shuffles on gfx1250).
#pragma unroll
        for (int off = 16; off > 0; off >>= 1) {
            unsigned long long o = __shfl_xor(key, off, 32);
            key = (o > key) ? o : key;
        }

        const int par = k & 1;
        if ((t & 31) == 0) wbest[par][t >> 5] = key;
        __syncthreads();   // single barrier per round (parity buffer -> no reset race)

        // Lane-parallel cross-wave combine: lane L loads entry (L & 15); the two
        // 16-lane halves hold identical copies, so a 4-step xor-shuffle max over
        // lane bits [3:0] converges every lane to the global max.
        unsigned long long red = wbest[par][t & (NWAVES - 1)];
#pragma unroll
        for (int off = NWAVES / 2; off > 0; off >>= 1) {
            unsigned long long o = __shfl_xor(red, off, 32);
            red = (o > red) ? o : red;
        }
        const unsigned idx = 0xFFFFFFFFu - (unsigned)(red & 0xFFFFFFFFull);

        // Broadcast winner coords from LDS (same address all lanes -> bank bcast).
        wx = pts[3 * idx + 0];
        wy = pts[3 * idx + 1];
        wz = pts[3 * idx + 2];

        if (t == 0) {
            float* o = xout + ((size_t)b * KSAMP + k) * 3;
            o[0] = wx; o[1] = wy; o[2] = wz;
        }
    }
}

// batch_out[j] = j / KSAMP  (int32 region after the B*K*3 float coords)
__global__ void batch_kernel(int* __restrict__ out) {
    const int j = blockIdx.x * blockDim.x + threadIdx.x;
    if (j < B_CLOUDS * KSAMP) out[j] = j >> 12;   // KSAMP = 4096 = 2^12
}

extern "C" void kernel_launch(void* const* d_in, const int* in_sizes, int n_in,
                              void* d_out, int out_size, void* d_ws, size_t ws_size,
                              hipStream_t stream) {
    (void)in_sizes; (void)n_in; (void)d_ws; (void)ws_size; (void)out_size;
    const float* x = (const float*)d_in[0];
    float* xout = (float*)d_out;                              // [B*K, 3] floats
    int* bout = (int*)(xout + (size_t)B_CLOUDS * KSAMP * 3);  // [B*K] int32

    fps_kernel<<<dim3(B_CLOUDS), dim3(TPB), 0, stream>>>(x, xout);

    const int bk = B_CLOUDS * KSAMP;
    batch_kernel<<<(bk + 255) / 256, 256, 0, stream>>>(bout);
}